// GCNNet_20383914786996
// MI455X (gfx1250) — compile-verified
//
#include <hip/hip_runtime.h>
#include <hip/hip_bf16.h>

typedef float v2f __attribute__((ext_vector_type(2)));
typedef float v8f __attribute__((ext_vector_type(8)));

#define D_H 128
// K-pair interleaved LDS layout: wt2[(k>>1)*PAIR_STRIDE + n*2 + (k&1)]
// PAIR_STRIDE = 288 dwords (256 + 32 pad) => consecutive k-pairs are offset by
// 32 banks, so the two half-wave K-groups read disjoint bank halves.
#define PAIR_STRIDE 288

// ---------------------------------------------------------------------------
// C[M x 128] = f_post( f_pre(A)[M x 128] @ W[128 x 128] )
//   f_pre : optional per-K bias (+ optional relu) applied to A on load
//   f_post: optional per-N bias (+ optional relu) applied to C on store
// One wave -> one 16(M) x 128(N) output strip: 8 accumulators, A read once.
// Whole W staged in LDS (64 k-pairs x 288 dwords = 72 KB). 4 waves / block.
// ---------------------------------------------------------------------------
template <bool PRE_BIAS, bool PRE_RELU, bool POST_BIAS, bool POST_RELU>
__global__ void gemm_wmma_f32(const float* __restrict__ A,
                              const float* __restrict__ W,
                              const float* __restrict__ preB,
                              const float* __restrict__ postB,
                              float* __restrict__ C, int M) {
    __shared__ float wt2[64 * PAIR_STRIDE];

    const int tid  = threadIdx.x;
    const int lane = tid & 31;
    const int wave = tid >> 5;

    // cooperative stage of full W into K-pair interleaved layout
    for (int idx = tid; idx < 128 * 128; idx += 128) {
        const int k = idx >> 7;
        const int n = idx & 127;
        wt2[(k >> 1) * PAIR_STRIDE + n * 2 + (k & 1)] = W[idx];
    }
    __syncthreads();

    const int mTile = blockIdx.x * 4 + wave;
    if (mTile * 16 >= M) return;          // wave-uniform: EXEC stays all-ones
    const int mBase = mTile * 16;

    const int half = lane >> 4;           // K sub-block select (0 or 1)
    const int l15  = lane & 15;
    const long long aRow = (long long)(mBase + l15) * D_H;

    v8f acc[8];
#pragma unroll
    for (int nt = 0; nt < 8; ++nt) acc[nt] = (v8f){0.f,0.f,0.f,0.f,0.f,0.f,0.f,0.f};

    for (int k0 = 0; k0 < 128; k0 += 4) {
        const int ka = k0 + half * 2;     // this lane's K pair: (ka, ka+1)
        // A: contiguous 8B load of A[row][ka..ka+1]
        const float2 av = *(const float2*)(A + aRow + ka);
        v2f a;
        if (PRE_BIAS) {
            float ax = av.x + preB[ka];
            float ay = av.y + preB[ka + 1];
            if (PRE_RELU) { ax = fmaxf(ax, 0.0f); ay = fmaxf(ay, 0.0f); }
            a.x = ax; a.y = ay;
        } else {
            a.x = av.x; a.y = av.y;
        }
        const float* bbase = wt2 + (ka >> 1) * PAIR_STRIDE;
#pragma unroll
        for (int nt = 0; nt < 8; ++nt) {
            // B: contiguous 8B LDS load of (W[ka][col], W[ka+1][col])
            const float2 bv = *(const float2*)(bbase + (nt * 16 + l15) * 2);
            v2f b; b.x = bv.x; b.y = bv.y;
            acc[nt] = __builtin_amdgcn_wmma_f32_16x16x4_f32(
                false, a, false, b, (short)0, acc[nt], false, false);
        }
    }

#pragma unroll
    for (int nt = 0; nt < 8; ++nt) {
        const int col = nt * 16 + l15;
        const float bv = POST_BIAS ? postB[col] : 0.0f;
#pragma unroll
        for (int r = 0; r < 8; ++r) {
            const int row = mBase + half * 8 + r;
            float v = acc[nt][r] + bv;
            if (POST_RELU) v = fmaxf(v, 0.0f);
            C[(long long)row * D_H + col] = v;
        }
    }
}

// ---------------------------------------------------------------------------
__global__ void fill_zero(float* __restrict__ p, long long n) {
    long long i = (long long)blockIdx.x * blockDim.x + threadIdx.x;
    if (i < n) p[i] = 0.0f;
}

// out[dst] += h[src] * ew  -- one thread per (edge, 4-dim chunk)
__global__ void scatter_edges(const float* __restrict__ h,
                              const int* __restrict__ src,
                              const int* __restrict__ dst,
                              const float* __restrict__ ew,
                              float* __restrict__ out, int nE) {
    long long idx = (long long)blockIdx.x * blockDim.x + threadIdx.x;
    long long total = (long long)nE * 32;
    if (idx >= total) return;
    const int e  = (int)(idx >> 5);
    const int d4 = (int)(idx & 31) << 2;
    const int s = src[e];
    const int t = dst[e];
    const float w = ew[e];
    const float4 v = *(const float4*)(h + (long long)s * D_H + d4);
    float* o = out + (long long)t * D_H + d4;
    atomicAdd(o + 0, v.x * w);
    atomicAdd(o + 1, v.y * w);
    atomicAdd(o + 2, v.z * w);
    atomicAdd(o + 3, v.w * w);
}

// g[n] = dot(t[n,:], gW2) + gb2 ; atomic segment-max via order-preserving uint
__global__ void gate_kernel(const float* __restrict__ t,
                            const float* __restrict__ gW2,
                            const float* __restrict__ gb2,
                            const int* __restrict__ batch,
                            float* __restrict__ g,
                            unsigned* __restrict__ m_enc, int N) {
    const int node = blockIdx.x * (blockDim.x >> 5) + (threadIdx.x >> 5);
    const int lane = threadIdx.x & 31;
    if (node >= N) return;
    const float* row = t + (long long)node * D_H;
    float s = 0.0f;
#pragma unroll
    for (int i = 0; i < 4; ++i) s += row[lane + 32 * i] * gW2[lane + 32 * i];
    for (int off = 16; off > 0; off >>= 1) s += __shfl_down(s, off);
    if (lane == 0) {
        const float gv = s + gb2[0];
        g[node] = gv;
        const int bits = __float_as_int(gv);
        const unsigned enc = (bits >= 0) ? ((unsigned)bits | 0x80000000u)
                                         : ~(unsigned)bits;
        atomicMax(m_enc + batch[node], enc);
    }
}

__global__ void exp_denom(float* __restrict__ g,
                          const unsigned* __restrict__ m_enc,
                          const int* __restrict__ batch,
                          float* __restrict__ denom, int N) {
    int n = blockIdx.x * blockDim.x + threadIdx.x;
    if (n >= N) return;
    const int b = batch[n];
    const unsigned u = m_enc[b];
    const int bits = (u & 0x80000000u) ? (int)(u ^ 0x80000000u) : (int)~u;
    const float m = __int_as_float(bits);
    const float e = __expf(g[n] - m);
    g[n] = e;
    atomicAdd(denom + b, e);
}

// p[batch[n],:] += alpha_n * (agg2[n,:] + b2)   (bias fused here)
__global__ void attn_pool(const float* __restrict__ agg2,
                          const float* __restrict__ b2,
                          const float* __restrict__ e,
                          const float* __restrict__ denom,
                          const int* __restrict__ batch,
                          float* __restrict__ p, int N) {
    long long idx = (long long)blockIdx.x * blockDim.x + threadIdx.x;
    if (idx >= (long long)N * D_H) return;
    const int n = (int)(idx >> 7);
    const int d = (int)(idx & (D_H - 1));
    const int b = batch[n];
    const float alpha = e[n] / (denom[b] + 1e-16f);
    atomicAdd(p + b * D_H + d, alpha * (agg2[idx] + b2[d]));
}

// out[g] = relu(p[g,:] @ l1W + l1b) @ l2W + l2b   (one block per graph)
__global__ void head_kernel(const float* __restrict__ p,
                            const float* __restrict__ l1W,
                            const float* __restrict__ l1b,
                            const float* __restrict__ l2W,
                            const float* __restrict__ l2b,
                            float* __restrict__ out) {
    __shared__ float row[D_H];
    __shared__ float q[D_H];
    const int g = blockIdx.x;
    const int d = threadIdx.x;
    row[d] = p[g * D_H + d];
    __syncthreads();
    float s = 0.0f;
    for (int k = 0; k < D_H; ++k) s += row[k] * l1W[k * D_H + d];
    q[d] = fmaxf(s + l1b[d], 0.0f) * l2W[d];
    __syncthreads();
    for (int off = 64; off > 0; off >>= 1) {
        if (d < off) q[d] += q[d + off];
        __syncthreads();
    }
    if (d == 0) out[g] = q[0] + l2b[0];
}

// ---------------------------------------------------------------------------
extern "C" void kernel_launch(void* const* d_in, const int* in_sizes, int n_in,
                              void* d_out, int out_size, void* d_ws, size_t ws_size,
                              hipStream_t stream) {
    const float* x   = (const float*)d_in[0];
    const int*   ei  = (const int*)  d_in[1];
    const float* ew  = (const float*)d_in[2];
    const int*   bat = (const int*)  d_in[3];
    const float* W1  = (const float*)d_in[4];
    const float* b1  = (const float*)d_in[5];
    const float* W2  = (const float*)d_in[6];
    const float* b2  = (const float*)d_in[7];
    const float* gW1 = (const float*)d_in[8];
    const float* gb1 = (const float*)d_in[9];
    const float* gW2 = (const float*)d_in[10];
    const float* gb2 = (const float*)d_in[11];
    const float* l1W = (const float*)d_in[12];
    const float* l1b = (const float*)d_in[13];
    const float* l2W = (const float*)d_in[14];
    const float* l2b = (const float*)d_in[15];
    float* out = (float*)d_out;

    const int N = in_sizes[0] / D_H;      // 100000
    const int E = in_sizes[2];            // 1600000
    const int G = out_size;               // 128
    const int* src = ei;
    const int* dst = ei + E;

    const long long NM = (long long)N * D_H;

    // workspace layout
    char* ws = (char*)d_ws;
    float*    bufA  = (float*)ws;                                  // N x 128
    float*    bufB  = (float*)(ws + NM * sizeof(float));           // N x 128
    float*    gate  = (float*)(ws + 2 * NM * sizeof(float));       // N
    unsigned* m_enc = (unsigned*)((char*)gate + ((size_t)N * 4 + 511) / 512 * 512);
    float*    denom = (float*)((char*)m_enc + 512);                // G
    float*    pmat  = (float*)((char*)denom + 512);                // G x 128
    (void)ws_size; (void)n_in;

    const int mTiles = (N + 15) / 16;                 // 6250
    dim3 gemmGrid((mTiles + 3) / 4);                  // 1563 blocks, 4 waves each
    dim3 gemmBlk(128);

    const int zBlk = 256;
    const int zGrdNM = (int)((NM + zBlk - 1) / zBlk);
    const long long scTot = (long long)E * 32;
    const int scGrd = (int)((scTot + 255) / 256);

    // ---- layer 1: agg1 = segsum((x@W1)[src]*ew -> dst) ----
    gemm_wmma_f32<false,false,false,false><<<gemmGrid, gemmBlk, 0, stream>>>(
        x, W1, nullptr, nullptr, bufA, N);
    fill_zero<<<zGrdNM, zBlk, 0, stream>>>(bufB, NM);
    scatter_edges<<<scGrd, 256, 0, stream>>>(bufA, src, dst, ew, bufB, E);

    // ---- layer 2: agg2 = segsum((relu(agg1+b1)@W2)[src]*ew -> dst) ----
    gemm_wmma_f32<true,true,false,false><<<gemmGrid, gemmBlk, 0, stream>>>(
        bufB, W2, b1, nullptr, bufA, N);
    fill_zero<<<zGrdNM, zBlk, 0, stream>>>(bufB, NM);
    scatter_edges<<<scGrd, 256, 0, stream>>>(bufA, src, dst, ew, bufB, E);
    // bufB = agg2  (h2 = agg2 + b2, bias applied on the fly downstream)

    // ---- attention pooling ----
    // t = relu((agg2+b2) @ gW1 + gb1) -> bufA
    gemm_wmma_f32<true,false,true,true><<<gemmGrid, gemmBlk, 0, stream>>>(
        bufB, gW1, b2, gb1, bufA, N);

    // zero m_enc (encoded -inf identity == 0), denom, pmat
    fill_zero<<<(256 + G * D_H + 255) / 256, 256, 0, stream>>>((float*)m_enc,
        256 + (long long)G * D_H);

    gate_kernel<<<(N * 32 + 255) / 256, 256, 0, stream>>>(bufA, gW2, gb2, bat, gate, m_enc, N);
    exp_denom<<<(N + 255) / 256, 256, 0, stream>>>(gate, m_enc, bat, denom, N);
    attn_pool<<<(int)((NM + 255) / 256), 256, 0, stream>>>(bufB, b2, gate, denom, bat, pmat, N);

    // ---- head ----
    head_kernel<<<G, D_H, 0, stream>>>(pmat, l1W, l1b, l2W, l2b, out);
}